// QuantizedKVCache_3015067042366
// MI455X (gfx1250) — compile-verified
//
#include <hip/hip_runtime.h>
#include <stdint.h>

// ---------------------------------------------------------------------------
// QuantizedKVCache for MI455X (gfx1250, wave32).
// Pure streaming workload (~332 MB/call -> ~14 us roofline @ 23.3 TB/s).
// No matrix math -> no WMMA. CDNA5 paths used instead:
//   * global_load_async_to_lds_b32 (ASYNCcnt) staging int8 rows into LDS,
//     16 copies in flight per wave, drained with progressive s_wait_asynccnt
//   * ds_load readback, 128-bit non-temporal stores (output > 192MB L2)
// ---------------------------------------------------------------------------

typedef float v4f __attribute__((ext_vector_type(4)));

// Builtin parameter address spaces (clang prints AS1 as "__device__").
typedef __attribute__((address_space(1))) int gas_int;   // global
typedef __attribute__((address_space(3))) int las_int;   // LDS

static constexpr int  CB = 2, CS_MAX = 4096, CH = 32, CD = 128, CS_IN = 128;
static constexpr long TOK_FULL       = (long)CB * CS_MAX * CH;   // 262144 per tensor
static constexpr long TOK_NEW        = (long)CB * CS_IN  * CH;   // 8192   per tensor
static constexpr long OUT_PER_TENSOR = TOK_FULL * CD;            // 33554432 floats
static constexpr int  TPW            = 16;                       // tokens per wave (B)
static constexpr float QMINF = -128.0f, QMAXF = 127.0f;
static constexpr float EPSF  = 1.1920929e-07f;                   // float32 eps

#if defined(__has_builtin)
#if __has_builtin(__builtin_amdgcn_global_load_async_to_lds_b32) && \
    __has_builtin(__builtin_amdgcn_s_wait_asynccnt)
#define HAVE_ASYNC_LDS 1
#else
#define HAVE_ASYNC_LDS 0
#endif
#else
#define HAVE_ASYNC_LDS 0
#endif

__device__ __forceinline__ int clamp_start(int st) {
  // jax.lax.dynamic_update_slice clamps the start index so the slice fits.
  st = st < 0 ? 0 : st;
  return st > (CS_MAX - CS_IN) ? (CS_MAX - CS_IN) : st;
}

// ---------------------------------------------------------------------------
// Kernel B: dequantize every token OUTSIDE the updated slice.
// One wave32 per 16 consecutive tokens (same (b,s) for all 16 since H=32, so
// the slice-skip predicate is wave-uniform). Software-pipelined via ASYNCcnt:
// issue 16 async 128B row copies, drain in groups of 4 with progressive waits.
// ---------------------------------------------------------------------------
__global__ void __launch_bounds__(256)
kv_dequant_cache(const int* __restrict__ input_pos,
                 const int8_t* __restrict__ k_cache, const int8_t* __restrict__ v_cache,
                 const float*  __restrict__ k_sc,    const float*  __restrict__ v_sc,
                 const int*    __restrict__ k_zp,    const int*    __restrict__ v_zp,
                 float* __restrict__ out) {
  const int  lane = threadIdx.x & 31;
  const long wid  = (((long)blockIdx.x * blockDim.x) + threadIdx.x) >> 5;
  if (wid >= (2 * TOK_FULL) / TPW) return;             // 32768 waves

  const int  start  = clamp_start(input_pos[0]);
  const long tbase  = wid * TPW;                       // first token id
  const int  tensor = (int)(tbase / TOK_FULL);
  const long rem0   = tbase - (long)tensor * TOK_FULL; // (b*S_MAX + s)*H + h
  const int  s      = (int)((rem0 / CH) % CS_MAX);
  if ((unsigned)(s - start) < (unsigned)CS_IN) return; // fresh tokens: kernel A

  const int8_t* cache = tensor ? v_cache : k_cache;
  const float*  scp   = tensor ? v_sc    : k_sc;
  const int*    zpp   = tensor ? v_zp    : k_zp;

  // Coalesced qparam fetch: lanes 0..15 carry token 0..15's scale/zp.
  const float sc_l = scp[rem0 + (lane & (TPW - 1))];
  const float zp_l = (float)zpp[rem0 + (lane & (TPW - 1))];

  const long  rowbase = rem0 * CD;                     // elem offset of token 0
  float* const outb   = out + (long)tensor * OUT_PER_TENSOR + rowbase;

#if HAVE_ASYNC_LDS
  __shared__ __align__(16) int8_t lbuf[8][TPW * CD];   // 8 waves * 2KB = 16KB
  int8_t* const       wb = lbuf[threadIdx.x >> 5];
  const int8_t* const gb = cache + rowbase;

  // Fill the pipe: 16 outstanding async row copies (128B each, lane-parallel).
#pragma unroll
  for (int t = 0; t < TPW; ++t) {
    __builtin_amdgcn_global_load_async_to_lds_b32(
        (gas_int*)(gb + t * CD + lane * 4),
        (las_int*)(wb + t * CD + lane * 4), /*imm offset=*/0, /*cpol=*/0);
  }

  auto proc = [&](int t) {
    const float scale = __shfl(sc_l, t, 32);
    const float zpf   = __shfl(zp_l, t, 32);
    const int   w     = *(const int*)(wb + t * CD + lane * 4);  // ds_load_b32
    v4f o;
    o.x = ((float)(int)(int8_t)( w       ) - zpf) * scale;
    o.y = ((float)(int)(int8_t)( w >>  8 ) - zpf) * scale;
    o.z = ((float)(int)(int8_t)( w >> 16 ) - zpf) * scale;
    o.w = ((float)(int)(int8_t)( w >> 24 ) - zpf) * scale;
    __builtin_nontemporal_store(o, (v4f*)(outb + t * CD) + lane);
  };

  // Progressive drain: process 4 rows while 12/8/4/0 copies remain in flight.
  __builtin_amdgcn_s_wait_asynccnt(12);
  __asm__ __volatile__("" ::: "memory");
  proc(0);  proc(1);  proc(2);  proc(3);
  __builtin_amdgcn_s_wait_asynccnt(8);
  __asm__ __volatile__("" ::: "memory");
  proc(4);  proc(5);  proc(6);  proc(7);
  __builtin_amdgcn_s_wait_asynccnt(4);
  __asm__ __volatile__("" ::: "memory");
  proc(8);  proc(9);  proc(10); proc(11);
  __builtin_amdgcn_s_wait_asynccnt(0);
  __asm__ __volatile__("" ::: "memory");
  proc(12); proc(13); proc(14); proc(15);
#else
  const int8_t* const gb = cache + rowbase;
#pragma unroll
  for (int t = 0; t < TPW; ++t) {
    const float scale = __shfl(sc_l, t, 32);
    const float zpf   = __shfl(zp_l, t, 32);
    const int   w     = *(const int*)(gb + t * CD + lane * 4);
    v4f o;
    o.x = ((float)(int)(int8_t)( w       ) - zpf) * scale;
    o.y = ((float)(int)(int8_t)( w >>  8 ) - zpf) * scale;
    o.z = ((float)(int)(int8_t)( w >> 16 ) - zpf) * scale;
    o.w = ((float)(int)(int8_t)( w >> 24 ) - zpf) * scale;
    __builtin_nontemporal_store(o, (v4f*)(outb + t * CD) + lane);
  }
#endif
}

// ---------------------------------------------------------------------------
// Kernel A: per-token asymmetric int8 quantize + immediate dequantize of the
// fresh k_val/v_val tokens; writes fp32 into the updated slice of the output.
// One wave32 per token; lane holds 4 floats (b128 load), 5-step shfl_xor
// butterfly for the wave-wide min/max (warpSize == 32 on gfx1250).
// ---------------------------------------------------------------------------
__global__ void __launch_bounds__(256)
kv_quant_new(const int* __restrict__ input_pos,
             const float* __restrict__ k_val, const float* __restrict__ v_val,
             float* __restrict__ out) {
  const int  lane = threadIdx.x & 31;
  const long wid  = (((long)blockIdx.x * blockDim.x) + threadIdx.x) >> 5;
  if (wid >= 2 * TOK_NEW) return;

  const int start  = clamp_start(input_pos[0]);
  const int tensor = (int)(wid / TOK_NEW);
  const long rem   = wid - (long)tensor * TOK_NEW;    // (b*S_IN + s)*H + h
  const int  h     = (int)(rem % CH);
  const int  s     = (int)((rem / CH) % CS_IN);
  const int  b     = (int)(rem / ((long)CH * CS_IN));

  const float* src = (tensor ? v_val : k_val) + rem * CD;
  const v4f x = *((const v4f*)src + lane);

  float mn = fminf(fminf(x.x, x.y), fminf(x.z, x.w));
  float mx = fmaxf(fmaxf(x.x, x.y), fmaxf(x.z, x.w));
#pragma unroll
  for (int m = 16; m >= 1; m >>= 1) {
    mn = fminf(mn, __shfl_xor(mn, m, 32));
    mx = fmaxf(mx, __shfl_xor(mx, m, 32));
  }
  mn = fminf(mn, 0.0f);
  mx = fmaxf(mx, 0.0f);

  const float scale = fmaxf((mx - mn) / 255.0f, EPSF);
  const float dmin  = mn / scale;
  const float dmax  = mx / scale;
  // Zero point minimizing endpoint error (matches torch decomposed impl).
  float zp_f = ((QMINF + dmin) + (QMAXF + dmax) > 0.0f) ? (QMINF - dmin)
                                                        : (QMAXF - dmax);
  const float zpf = fminf(fmaxf(__builtin_rintf(zp_f), QMINF), QMAXF);

  v4f o;
  {
    float q;
    q = fminf(fmaxf(__builtin_rintf(x.x / scale) + zpf, QMINF), QMAXF);
    o.x = (q - zpf) * scale;
    q = fminf(fmaxf(__builtin_rintf(x.y / scale) + zpf, QMINF), QMAXF);
    o.y = (q - zpf) * scale;
    q = fminf(fmaxf(__builtin_rintf(x.z / scale) + zpf, QMINF), QMAXF);
    o.z = (q - zpf) * scale;
    q = fminf(fmaxf(__builtin_rintf(x.w / scale) + zpf, QMINF), QMAXF);
    o.w = (q - zpf) * scale;
  }

  const long orow = (((long)b * CS_MAX + start + s) * CH + h) * CD;
  v4f* dst = (v4f*)(out + (long)tensor * OUT_PER_TENSOR + orow) + lane;
  __builtin_nontemporal_store(o, dst);
}

// ---------------------------------------------------------------------------
// Inputs (setup_inputs order):
//  0 input_pos i32[1]            1 k_val f32[2,128,32,128]  2 v_val f32[...]
//  3 k_cache i8[2,4096,32,128]   4 v_cache i8[...]
//  5 k_scales f32[2,4096,32,1]   6 v_scales f32[...]
//  7 k_zp i32[2,4096,32,1]       8 v_zp i32[...]
// Output: k_out fp32 (33554432) then v_out fp32 (33554432), flat.
// ---------------------------------------------------------------------------
extern "C" void kernel_launch(void* const* d_in, const int* in_sizes, int n_in,
                              void* d_out, int out_size, void* d_ws, size_t ws_size,
                              hipStream_t stream) {
  (void)in_sizes; (void)n_in; (void)out_size; (void)d_ws; (void)ws_size;
  const int*    input_pos = (const int*)   d_in[0];
  const float*  k_val     = (const float*) d_in[1];
  const float*  v_val     = (const float*) d_in[2];
  const int8_t* k_cache   = (const int8_t*)d_in[3];
  const int8_t* v_cache   = (const int8_t*)d_in[4];
  const float*  k_sc      = (const float*) d_in[5];
  const float*  v_sc      = (const float*) d_in[6];
  const int*    k_zp      = (const int*)   d_in[7];
  const int*    v_zp      = (const int*)   d_in[8];
  float*        out       = (float*)       d_out;

  // Kernel B: 32768 waves * 32 = 1,048,576 threads -> 4096 blocks of 256.
  kv_dequant_cache<<<4096, 256, 0, stream>>>(input_pos, k_cache, v_cache,
                                             k_sc, v_sc, k_zp, v_zp, out);
  // Kernel A: 2*8192 waves = 524,288 threads -> 2048 blocks of 256.
  kv_quant_new<<<2048, 256, 0, stream>>>(input_pos, k_val, v_val, out);
}